// GuidedFilter_8615704396458
// MI455X (gfx1250) — compile-verified
//
#include <hip/hip_runtime.h>
#include <hip/hip_bf16.h>

// Guided filter, fp32, via V_WMMA_F32_16X16X4_F32 banded-matrix box sums.
// Stage 1: box sums of x, y, x*y, x*x  -> per-pixel A, b (written to d_ws).
// Stage 2: box sums of A, b           -> out = meanA * x + meanB.
// Box sums use zero padding; normalization uses the analytic valid-window
// count, which is exactly box(ones)-normalization as in the reference.
// Halo staging into LDS uses GLOBAL_LOAD_ASYNC_TO_LDS_B128 (ASYNCcnt) for
// fully-interior tiles; border tiles use a predicated zero-fill path.

typedef float v2f __attribute__((ext_vector_type(2)));
typedef float v8f __attribute__((ext_vector_type(8)));

#define TS    16          // WMMA tile edge
#define RAD   5           // box radius (reference r=5)
#define OT    64          // output tile edge per workgroup
#define XT    96          // OT + 2*TS : LDS halo region edge
#define XP    100         // padded pitch (floats) for input regions (bank-conflict avoidance)
#define HP    68          // padded pitch (floats) for horizontal-sum buffer
#define IMH   512
#define IMW   512
#define NIMG  24          // B*C = 8*3

// D(16x16,f32) = A(16x4,f32) x B(4x16,f32) + C
__device__ __forceinline__ v8f wmma4(v2f a, v2f b, v8f c) {
  return __builtin_amdgcn_wmma_f32_16x16x4_f32(
      /*neg_a=*/false, a, /*neg_b=*/false, b,
      /*c_mod=*/(short)0, c, /*reuse_a=*/false, /*reuse_b=*/false);
}

// Banded 0/1 matrix element: input position (16p + k) in the 48-wide strip,
// output position mn in the center 16; 1 iff |(16p + k) - 16 - mn| <= RAD.
__device__ __forceinline__ float band(int kk, int mn) {
  int d = kk - TS - mn;
  return (d >= -RAD && d <= RAD) ? 1.0f : 0.0f;
}

// Async DMA of 16 bytes global -> LDS (no VGPR staging). Generic pointers to
// __shared__ carry the LDS byte offset in their low 32 bits;
// the instruction adds the wave's LDS_BASE itself.
__device__ __forceinline__ void async_copy16(void* lds_dst, const float* gsrc) {
  uint32_t lds_off = (uint32_t)(uintptr_t)lds_dst;
  asm volatile("global_load_async_to_lds_b128 %0, %1, off"
               :: "v"(lds_off), "v"(gsrc)
               : "memory");
}

__device__ __forceinline__ void async_wait0() {
  asm volatile("s_wait_asynccnt 0" ::: "memory");
}

// Load a 96x96 halo region of two planes into LDS.
// interior==true: whole region in-bounds -> async LDS-direct 16B DMA copies.
// interior==false: per-element predicated loads with zero fill.
__device__ __forceinline__ void load_region(const float* __restrict__ p0,
                                            const float* __restrict__ p1,
                                            float* s0, float* s1,
                                            int R0, int C0, int tid, bool interior) {
  if (interior) {
    // 96 rows x 24 chunks of 4 floats = 2304 chunks; 256 threads x 9 iters.
    for (int i = tid; i < XT * (XT / 4); i += 256) {
      int ly = i / (XT / 4);
      int lx = (i - ly * (XT / 4)) * 4;
      int off = (R0 - TS + ly) * IMW + (C0 - TS + lx);
      int so  = ly * XP + lx;
      async_copy16(&s0[so], p0 + off);
      async_copy16(&s1[so], p1 + off);
    }
    async_wait0();
  } else {
    for (int i = tid; i < XT * XT; i += 256) {
      int ly = i / XT;
      int lx = i - ly * XT;
      int gr = R0 - TS + ly;
      int gc = C0 - TS + lx;
      float a = 0.0f, b = 0.0f;
      if (gr >= 0 && gr < IMH && gc >= 0 && gc < IMW) {
        int off = gr * IMW + gc;
        a = p0[off];
        b = p1[off];
      }
      s0[ly * XP + lx] = a;
      s1[ly * XP + lx] = b;
    }
  }
}

// Horizontal 11-tap sliding sums for quantity Q over the whole 96-row x 64-col
// region, as 16x16 tile products with banded matrices.
// Q: 0 -> sX, 1 -> sY, 2 -> sX*sY, 3 -> sX*sX.
template <int Q>
__device__ __forceinline__ void hsum_stage(const float* __restrict__ sX,
                                           const float* __restrict__ sY,
                                           float* __restrict__ sH,
                                           int wave, int lane) {
  const int m  = lane & 15;            // row within tile (A row id)
  const int n  = m;                    // col within tile (B/C/D col id)
  const int kb = (lane >> 4) << 1;     // K sub-offset: 0 (lanes 0-15) or 2
  const int hs = (lane >> 4) << 3;     // C/D row offset: 0 or 8
  // 24 Hsum tiles: 6 row-tiles x 4 col-tiles; 8 waves -> 3 tiles each.
  for (int t = wave; t < 24; t += 8) {
    const int ti = t >> 2;
    const int tj = t & 3;
    v8f acc = {};
#pragma unroll
    for (int p = 0; p < 3; ++p) {
      const float* rx = sX + (TS * ti + m) * XP + TS * (tj + p);
      const float* ry = sY + (TS * ti + m) * XP + TS * (tj + p);
#pragma unroll
      for (int c = 0; c < 4; ++c) {
        const int k0 = 4 * c + kb;
        v2f a, b;
        if (Q == 0)      { a[0] = rx[k0];            a[1] = rx[k0 + 1]; }
        else if (Q == 1) { a[0] = ry[k0];            a[1] = ry[k0 + 1]; }
        else if (Q == 2) { a[0] = rx[k0] * ry[k0];   a[1] = rx[k0 + 1] * ry[k0 + 1]; }
        else             { a[0] = rx[k0] * rx[k0];   a[1] = rx[k0 + 1] * rx[k0 + 1]; }
        b[0] = band(16 * p + k0,     n);
        b[1] = band(16 * p + k0 + 1, n);
        acc = wmma4(a, b, acc);
      }
    }
    const int col = TS * tj + n;
#pragma unroll
    for (int v = 0; v < 8; ++v) {
      sH[(TS * ti + v + hs) * HP + col] = acc[v];
    }
  }
}

// Vertical 11-tap sliding sums for one 16x16 output tile:
// D = sum_p T_p(16x16 banded) x Hsum(local row-tile oi+p, col-tile oj).
__device__ __forceinline__ v8f vsum_tile(const float* __restrict__ sH,
                                         int oi, int oj, int lane) {
  const int m  = lane & 15;
  const int n  = m;
  const int kb = (lane >> 4) << 1;
  v8f acc = {};
#pragma unroll
  for (int p = 0; p < 3; ++p) {
#pragma unroll
    for (int c = 0; c < 4; ++c) {
      const int k0 = 4 * c + kb;
      v2f a, b;
      a[0] = band(16 * p + k0,     m);
      a[1] = band(16 * p + k0 + 1, m);
      const float* h0 = sH + (TS * (oi + p) + k0) * HP + TS * oj + n;
      b[0] = h0[0];
      b[1] = h0[HP];
      acc = wmma4(a, b, acc);
    }
  }
  return acc;
}

__device__ __forceinline__ float window_inv_count(int gr, int gc) {
  int ch = min(gr + RAD, IMH - 1) - max(gr - RAD, 0) + 1;
  int cw = min(gc + RAD, IMW - 1) - max(gc - RAD, 0) + 1;
  return 1.0f / (float)(ch * cw);
}

__global__ __launch_bounds__(256)
void gf_stage1(const float* __restrict__ x, const float* __restrict__ y,
               float* __restrict__ outA, float* __restrict__ outB) {
  __shared__ float sX[XT * XP];
  __shared__ float sY[XT * XP];
  __shared__ float sH[XT * HP];
  const int tid  = threadIdx.x;
  const int wave = tid >> 5;
  const int lane = tid & 31;
  const int R0 = blockIdx.y * OT;
  const int C0 = blockIdx.x * OT;
  const bool interior = (R0 >= TS) && (R0 + OT + TS <= IMH) &&
                        (C0 >= TS) && (C0 + OT + TS <= IMW);
  const size_t ib = (size_t)blockIdx.z * (size_t)(IMH * IMW);

  load_region(x + ib, y + ib, sX, sY, R0, C0, tid, interior);
  __syncthreads();

  const int oi0 = wave >> 2;   // 0..1
  const int oj  = wave & 3;    // 0..3
  v8f S[4][2];

  hsum_stage<0>(sX, sY, sH, wave, lane);
  __syncthreads();
  S[0][0] = vsum_tile(sH, oi0,     oj, lane);
  S[0][1] = vsum_tile(sH, oi0 + 2, oj, lane);
  __syncthreads();

  hsum_stage<1>(sX, sY, sH, wave, lane);
  __syncthreads();
  S[1][0] = vsum_tile(sH, oi0,     oj, lane);
  S[1][1] = vsum_tile(sH, oi0 + 2, oj, lane);
  __syncthreads();

  hsum_stage<2>(sX, sY, sH, wave, lane);
  __syncthreads();
  S[2][0] = vsum_tile(sH, oi0,     oj, lane);
  S[2][1] = vsum_tile(sH, oi0 + 2, oj, lane);
  __syncthreads();

  hsum_stage<3>(sX, sY, sH, wave, lane);
  __syncthreads();
  S[3][0] = vsum_tile(sH, oi0,     oj, lane);
  S[3][1] = vsum_tile(sH, oi0 + 2, oj, lane);

  const int hs = (lane >> 4) << 3;
  const int n  = lane & 15;
#pragma unroll
  for (int st = 0; st < 2; ++st) {
    const int oi = oi0 + st * 2;
#pragma unroll
    for (int v = 0; v < 8; ++v) {
      const int gr = R0 + TS * oi + v + hs;
      const int gc = C0 + TS * oj + n;
      const float inv = window_inv_count(gr, gc);
      const float mx  = S[0][st][v] * inv;
      const float my  = S[1][st][v] * inv;
      const float cov = S[2][st][v] * inv - mx * my;
      const float var = S[3][st][v] * inv - mx * mx;
      const float a   = cov / (var + 0.01f);
      const float b   = my - a * mx;
      const size_t pix = ib + (size_t)gr * IMW + gc;
      outA[pix] = a;
      outB[pix] = b;
    }
  }
}

__global__ __launch_bounds__(256)
void gf_stage2(const float* __restrict__ A, const float* __restrict__ Bc,
               const float* __restrict__ x, float* __restrict__ out) {
  __shared__ float sX[XT * XP];
  __shared__ float sY[XT * XP];
  __shared__ float sH[XT * HP];
  const int tid  = threadIdx.x;
  const int wave = tid >> 5;
  const int lane = tid & 31;
  const int R0 = blockIdx.y * OT;
  const int C0 = blockIdx.x * OT;
  const bool interior = (R0 >= TS) && (R0 + OT + TS <= IMH) &&
                        (C0 >= TS) && (C0 + OT + TS <= IMW);
  const size_t ib = (size_t)blockIdx.z * (size_t)(IMH * IMW);

  load_region(A + ib, Bc + ib, sX, sY, R0, C0, tid, interior);
  __syncthreads();

  const int oi0 = wave >> 2;
  const int oj  = wave & 3;
  v8f SA[2], SB[2];

  hsum_stage<0>(sX, sY, sH, wave, lane);
  __syncthreads();
  SA[0] = vsum_tile(sH, oi0,     oj, lane);
  SA[1] = vsum_tile(sH, oi0 + 2, oj, lane);
  __syncthreads();

  hsum_stage<1>(sX, sY, sH, wave, lane);
  __syncthreads();
  SB[0] = vsum_tile(sH, oi0,     oj, lane);
  SB[1] = vsum_tile(sH, oi0 + 2, oj, lane);

  const int hs = (lane >> 4) << 3;
  const int n  = lane & 15;
#pragma unroll
  for (int st = 0; st < 2; ++st) {
    const int oi = oi0 + st * 2;
#pragma unroll
    for (int v = 0; v < 8; ++v) {
      const int gr = R0 + TS * oi + v + hs;
      const int gc = C0 + TS * oj + n;
      const float inv = window_inv_count(gr, gc);
      const size_t pix = ib + (size_t)gr * IMW + gc;
      const float meanA = SA[st][v] * inv;
      const float meanB = SB[st][v] * inv;
      out[pix] = meanA * x[pix] + meanB;
    }
  }
}

extern "C" void kernel_launch(void* const* d_in, const int* in_sizes, int n_in,
                              void* d_out, int out_size, void* d_ws, size_t ws_size,
                              hipStream_t stream) {
  const float* x = (const float*)d_in[0];
  const float* y = (const float*)d_in[1];
  // d_in[2] is r (==5), baked into RAD.
  float* out = (float*)d_out;

  const size_t npix = (size_t)NIMG * IMH * IMW;   // 6,291,456
  float* wsA = (float*)d_ws;
  float* wsB = wsA + npix;                         // needs 2*npix*4 = ~50.4 MB of d_ws

  dim3 grid(IMW / OT, IMH / OT, NIMG);             // 8 x 8 x 24
  dim3 block(256);
  gf_stage1<<<grid, block, 0, stream>>>(x, y, wsA, wsB);
  gf_stage2<<<grid, block, 0, stream>>>(wsA, wsB, x, out);
}